// HyperRelationLearner_20976620274287
// MI455X (gfx1250) — compile-verified
//
#include <hip/hip_runtime.h>

// Problem constants (match reference)
#define NUM_ENTITY 100000
#define DIM 200
#define HALF_DIM 100
#define N_QUALS 500000
#define N_QUERIES 200000
#define ALPHA 0.8f

typedef __attribute__((ext_vector_type(2))) float v2f;
typedef __attribute__((ext_vector_type(8))) float v8f;

#define LDS_STRIDE 204   // 200 + 4 pad: 204 % 64 = 12 -> bank conflict only at dM=16 (never hit)
#define ROWS_PER_BLOCK 16

__device__ __forceinline__ int lower_bound_i32(const int* __restrict__ arr, int n, int key) {
    int lo = 0, hi = n;
    while (lo < hi) {
        int mid = (lo + hi) >> 1;
        if (arr[mid] < key) lo = mid + 1; else hi = mid;
    }
    return lo;
}

// One-time 200x200 transpose of w_q into workspace: wqT[n][k] = wq[k][n].
// Makes B fragments (fixed n, consecutive k) contiguous -> global_load_b64.
__global__ __launch_bounds__(256)
void transpose_wq_kernel(const float* __restrict__ wq, float* __restrict__ wqT) {
    int idx = blockIdx.x * blockDim.x + threadIdx.x;
    if (idx < DIM * DIM) {
        int k = idx / DIM;
        int n = idx - k * DIM;
        wqT[n * DIM + k] = wq[k * DIM + n];
    }
}

__global__ __launch_bounds__(128)
void hyper_rel_fused_kernel(const float* __restrict__ ent,      // (NUM_ENTITY, DIM)
                            const float* __restrict__ rel,      // (2*NUM_REL, DIM)
                            const float* __restrict__ qrel,     // (NUM_QUAL_REL+1, DIM)
                            const float* __restrict__ wqT,      // (DIM, DIM) transposed
                            const int*   __restrict__ q_rel_idx,// (N_QUALS)
                            const int*   __restrict__ q_ent_idx,// (N_QUALS)
                            const int*   __restrict__ q_index,  // (N_QUALS) sorted
                            const int*   __restrict__ r_index,  // (N_QUERIES, 2)
                            float*       __restrict__ out)      // (N_QUERIES, DIM)
{
    __shared__ float tile[ROWS_PER_BLOCK * LDS_STRIDE];

    const int qbase = blockIdx.x * ROWS_PER_BLOCK;
    const int tid   = threadIdx.x;
    const int lane  = tid & 31;
    const int wave  = tid >> 5;

    // ---- Phase 1: build coalesced tile = segment_sum(rotate(q_ent, q_rel)) in LDS ----
    for (int rloc = wave * 4; rloc < wave * 4 + 4; ++rloc) {
        const int q  = qbase + rloc;
        const int lo = lower_bound_i32(q_index, N_QUALS, q);       // uniform per row
        const int hi = lower_bound_i32(q_index, N_QUALS, q + 1);
        for (int c = lane; c < HALF_DIM; c += 32) {
            float are = 0.0f, aim = 0.0f;
            for (int j = lo; j < hi; ++j) {
                const int e  = q_ent_idx[j];
                const int rr = q_rel_idx[j];
                const float hre = ent[e * DIM + c];
                const float him = ent[e * DIM + HALF_DIM + c];
                const float rre = qrel[rr * DIM + c];
                const float rim = qrel[rr * DIM + HALF_DIM + c];
                are += hre * rre - him * rim;   // real part
                aim += hre * rim + him * rre;   // imag part
            }
            tile[rloc * LDS_STRIDE + c]            = are;
            tile[rloc * LDS_STRIDE + HALF_DIM + c] = aim;
        }
    }
    __syncthreads();

    // ---- Phase 2: out = ALPHA*rel_part + (1-ALPHA)*(tile @ wq), f32 WMMA 16x16x4 ----
    // 13 N-tiles of 16 cols: 6 unmasked PAIRS (tiles 0..11) + 1 masked tile (12).
    // Pairing lets one A fragment (ds_load_b64) feed two WMMAs.
    const int mrow = lane & 15;        // A-matrix row / B,C,D column within tile
    const int koff = (lane >> 4) * 2;  // lanes 16-31 hold K+2,K+3 (ISA 16x4 f32 A layout)

    for (int p = wave; p < 7; p += 4) {
        if (p < 6) {
            // ---- hot path: two full tiles, no masking ----
            const int n0 = (2 * p) * 16 + mrow;
            const int n1 = n0 + 16;
            __builtin_prefetch(&wqT[n0 * DIM], 0, 1);   // global_prefetch_b8
            __builtin_prefetch(&wqT[n1 * DIM], 0, 1);

            v8f acc0 = {}, acc1 = {};
            for (int kb = 0; kb < DIM; kb += 4) {
                const int kk = kb + koff;
                const v2f a  = *(const v2f*)&tile[mrow * LDS_STRIDE + kk]; // ds_load_b64
                const v2f b0 = *(const v2f*)&wqT[n0 * DIM + kk];           // global_load_b64
                const v2f b1 = *(const v2f*)&wqT[n1 * DIM + kk];
                acc0 = __builtin_amdgcn_wmma_f32_16x16x4_f32(
                    false, a, false, b0, (short)0, acc0, false, false);
                acc1 = __builtin_amdgcn_wmma_f32_16x16x4_f32(
                    false, a, false, b1, (short)0, acc1, false, false);
            }
            #pragma unroll
            for (int i = 0; i < 8; ++i) {
                const int m  = i + (lane >> 4) * 8;   // C/D layout: vgpr i -> row i or i+8
                const int q  = qbase + m;
                const int ri = r_index[q * 2];        // r_index[q][0]
                out[q * DIM + n0] = ALPHA * rel[ri * DIM + n0] + (1.0f - ALPHA) * acc0[i];
                out[q * DIM + n1] = ALPHA * rel[ri * DIM + n1] + (1.0f - ALPHA) * acc1[i];
            }
        } else {
            // ---- remainder tile 12: cols 192..207, mask cols >= 200 ----
            const int n     = 192 + mrow;
            const int nc    = (n < DIM) ? n : (DIM - 1);   // clamped address, safe load
            const float msk = (n < DIM) ? 1.0f : 0.0f;     // zero OOB cols, EXEC stays full

            v8f acc = {};
            for (int kb = 0; kb < DIM; kb += 4) {
                const int kk = kb + koff;
                const v2f a = *(const v2f*)&tile[mrow * LDS_STRIDE + kk];
                v2f b = *(const v2f*)&wqT[nc * DIM + kk];
                b.x *= msk; b.y *= msk;
                acc = __builtin_amdgcn_wmma_f32_16x16x4_f32(
                    false, a, false, b, (short)0, acc, false, false);
            }
            if (n < DIM) {
                #pragma unroll
                for (int i = 0; i < 8; ++i) {
                    const int m  = i + (lane >> 4) * 8;
                    const int q  = qbase + m;
                    const int ri = r_index[q * 2];
                    out[q * DIM + n] = ALPHA * rel[ri * DIM + n] + (1.0f - ALPHA) * acc[i];
                }
            }
        }
    }
}

extern "C" void kernel_launch(void* const* d_in, const int* in_sizes, int n_in,
                              void* d_out, int out_size, void* d_ws, size_t ws_size,
                              hipStream_t stream) {
    const float* ent       = (const float*)d_in[0];
    const float* rel       = (const float*)d_in[1];
    const float* qrel      = (const float*)d_in[2];
    const float* wq        = (const float*)d_in[3];
    const int*   q_rel_idx = (const int*)d_in[4];
    const int*   q_ent_idx = (const int*)d_in[5];
    const int*   q_index   = (const int*)d_in[6];
    const int*   r_index   = (const int*)d_in[7];

    float* out_query = (float*)d_out;                         // (200000, 200)
    float* out_x     = out_query + (size_t)N_QUERIES * DIM;   // ent passthrough
    float* out_r     = out_x + (size_t)NUM_ENTITY * DIM;      // rel passthrough

    float* wqT = (float*)d_ws;                                // 200*200*4 = 160 KB scratch

    // Tuple outputs 2 & 3 are identity passthroughs -> D2D async copies
    hipMemcpyAsync(out_x, ent, (size_t)NUM_ENTITY * DIM * sizeof(float),
                   hipMemcpyDeviceToDevice, stream);
    hipMemcpyAsync(out_r, rel, (size_t)400 * DIM * sizeof(float),
                   hipMemcpyDeviceToDevice, stream);

    // One-time transpose of w_q so B fragments are contiguous b64 loads
    transpose_wq_kernel<<<(DIM * DIM + 255) / 256, 256, 0, stream>>>(wq, wqT);

    dim3 grid(N_QUERIES / ROWS_PER_BLOCK);   // 12500 blocks
    dim3 block(128);                          // 4 waves of 32
    hyper_rel_fused_kernel<<<grid, block, 0, stream>>>(
        ent, rel, qrel, wqT, q_rel_idx, q_ent_idx, q_index, r_index, out_query);
}